// SelfFeatureExtractor_37014028157612
// MI455X (gfx1250) — compile-verified
//
#include <hip/hip_runtime.h>

// ---------------------------------------------------------------------------
// out = relu( diag(1/(rowsum(adj)+1) + 1) * x @ W^T )
//   adj: [16384, 16384] f32, x: [16384, 512] f32, W: [256, 512] f32
//   out: [16384, 256] f32
//
// Phase 1 (bandwidth-bound, ~1 GiB @ 23.3 TB/s ~= 46 us): row-sum of adj,
//   d[row] = 1/(sum+1)+1, stored in d_ws.
// Phase 2: WMMA f32 GEMM (x @ W^T), epilogue applies d-row-scale + ReLU.
//   d is a row scale so it commutes with the K-reduction -> fuse at the end.
// ---------------------------------------------------------------------------

typedef __attribute__((ext_vector_type(2))) float v2f;
typedef __attribute__((ext_vector_type(4))) float v4f;
typedef __attribute__((ext_vector_type(8))) float v8f;

#define NROWS 16384
#define KDIM  512
#define EDIM  256

// ---------------- Phase 1: row-sum of adj -> d vector ----------------------
// One 256-thread block (8 waves) per row; each thread streams float4s.
__global__ __launch_bounds__(256) void rowsum_d_kernel(
    const float* __restrict__ adj, float* __restrict__ dvec) {
  const int row = blockIdx.x;
  const float* p = adj + (size_t)row * NROWS;

  float sum = 0.0f;
  // 16384 floats / row = 4096 float4; 256 threads -> 16 float4 each (b128 loads)
  for (int i = threadIdx.x * 4; i < NROWS; i += 256 * 4) {
    v4f v = *(const v4f*)(p + i);
    sum += (v[0] + v[1]) + (v[2] + v[3]);
  }

  // wave32 reduction
  #pragma unroll
  for (int off = 16; off > 0; off >>= 1)
    sum += __shfl_xor(sum, off, 32);

  __shared__ float partial[8];
  const int lane = threadIdx.x & 31;
  const int wave = threadIdx.x >> 5;
  if (lane == 0) partial[wave] = sum;
  __syncthreads();

  if (threadIdx.x == 0) {
    float s = 0.0f;
    #pragma unroll
    for (int w = 0; w < 8; ++w) s += partial[w];
    dvec[row] = 1.0f / (s + 1.0f) + 1.0f;
  }
}

// ---------------- Phase 2: WMMA f32 GEMM + d-scale + ReLU ------------------
// One wave computes one 16x16 output tile. K looped in steps of 4 using
// V_WMMA_F32_16X16X4_F32 (full fp32, matches reference precision).
//
// A (16x4 f32) per-lane layout: M = lane%16; VGPR0/1 hold K = 2*(lane/16)+{0,1}
// B (4x16 f32): mirrored striping -> lane supplies W[ecol = lane%16][same Ks]
// C/D (16x16 f32): VGPR r holds M = r + 8*(lane/16), N = lane%16
__global__ __launch_bounds__(256) void wmma_gemm_relu_kernel(
    const float* __restrict__ x, const float* __restrict__ W,
    const float* __restrict__ dvec, float* __restrict__ out) {
  const int lane   = threadIdx.x & 31;
  const int waveIb = threadIdx.x >> 5;
  const int waveId = blockIdx.x * 8 + waveIb;   // 16384 waves total

  const int tile_m = (waveId >> 4) * 16;        // 1024 M-tiles
  const int tile_e = (waveId & 15) * 16;        // 16 E-tiles

  const int mrow = lane & 15;                   // A row / B col / C col
  const int kgrp = lane >> 4;                   // selects K pair within step

  const float* arow = x + (size_t)(tile_m + mrow) * KDIM + 2 * kgrp;
  const float* brow = W + (size_t)(tile_e + mrow) * KDIM + 2 * kgrp;

  v8f acc0 = {};
  v8f acc1 = {};

  // 512 / 4 = 128 WMMA steps; two accumulators break the serial C-chain.
  #pragma unroll 8
  for (int k0 = 0; k0 < KDIM; k0 += 8) {
    v2f a0 = *(const v2f*)(arow + k0);
    v2f b0 = *(const v2f*)(brow + k0);
    v2f a1 = *(const v2f*)(arow + k0 + 4);
    v2f b1 = *(const v2f*)(brow + k0 + 4);
    acc0 = __builtin_amdgcn_wmma_f32_16x16x4_f32(
        false, a0, false, b0, (short)0, acc0, false, false);
    acc1 = __builtin_amdgcn_wmma_f32_16x16x4_f32(
        false, a1, false, b1, (short)0, acc1, false, false);
  }

  // Epilogue: combine accumulators, apply d-row scale + ReLU, store.
  const int crow0 = tile_m + kgrp * 8;          // first of this lane's 8 M rows
  #pragma unroll
  for (int r = 0; r < 8; ++r) {
    const int m = crow0 + r;
    float v = (acc0[r] + acc1[r]) * dvec[m];
    v = v > 0.0f ? v : 0.0f;
    out[(size_t)m * EDIM + tile_e + mrow] = v;
  }
}

// ---------------------------------------------------------------------------
extern "C" void kernel_launch(void* const* d_in, const int* in_sizes, int n_in,
                              void* d_out, int out_size, void* d_ws, size_t ws_size,
                              hipStream_t stream) {
  const float* adj = (const float*)d_in[0];   // [16384, 16384]
  const float* x   = (const float*)d_in[1];   // [16384, 512]
  const float* W   = (const float*)d_in[2];   // [256, 512]
  float* out  = (float*)d_out;                // [16384, 256]
  float* dvec = (float*)d_ws;                 // [16384] scratch

  // Phase 1: 16384 blocks x 256 threads, one block per adj row.
  rowsum_d_kernel<<<NROWS, 256, 0, stream>>>(adj, dvec);

  // Phase 2: 16384 tiles, 8 waves (256 threads) per block -> 2048 blocks.
  wmma_gemm_relu_kernel<<<2048, 256, 0, stream>>>(x, W, dvec, out);
}